// Sampler_90314572301058
// MI455X (gfx1250) — compile-verified
//
#include <hip/hip_runtime.h>
#include <stdint.h>
#include <limits.h>

// Problem constants (match reference).
#define BROWS     128
#define VV        152064
#define NTHREADS  512
#define CHUNK     297                 // VV / NTHREADS exactly
#define TILE_EL   (NTHREADS * 4)      // 2048 floats per async tile (b128/lane)
#define NTILES    74                  // 74*2048 = 151552
#define TAIL_BASE (NTILES * TILE_EL)  // + 512 tail elements = 152064
#define NREP      2                   // histogram replicas (contention relief)
#define CAND_CAP  4096                // candidate list capacity (reuses tile)

// Monotonic float -> uint mapping (total order matching float order).
__device__ __forceinline__ uint32_t f2ord(float f) {
  uint32_t u = __float_as_uint(f);
  return (u & 0x80000000u) ? ~u : (u | 0x80000000u);
}

// wave32 + block max reduction (16 waves of 32 lanes).
__device__ __forceinline__ float blockMax(float v, float* sred) {
  #pragma unroll
  for (int m = 16; m >= 1; m >>= 1) v = fmaxf(v, __shfl_xor(v, m, 32));
  int wave = threadIdx.x >> 5, lane = threadIdx.x & 31;
  if (lane == 0) sred[wave] = v;
  __syncthreads();
  if (threadIdx.x < 16) {
    v = sred[threadIdx.x];
    #pragma unroll
    for (int m = 8; m >= 1; m >>= 1) v = fmaxf(v, __shfl_xor(v, m, 32));
    if (threadIdx.x == 0) sred[0] = v;
  }
  __syncthreads();
  float r = sred[0];
  __syncthreads();
  return r;
}

// Parallel descending k-th selection over a (replicated) 4096-bin histogram.
// Each thread owns 8 bins; block suffix-scan finds the owner; owner resolves
// the exact bin and the residual k. Replaces a 4096-iteration serial loop.
__device__ __forceinline__ void selectBin(uint32_t (*hist)[4096], int K,
                                          uint32_t* sScanU,
                                          uint32_t* sBin, int* sKrem) {
  const int tid = threadIdx.x;
  uint32_t c[8]; uint32_t S = 0;
  #pragma unroll
  for (int b = 0; b < 8; ++b) {
    uint32_t cb = 0;
    #pragma unroll
    for (int r = 0; r < NREP; ++r) cb += hist[r][tid * 8 + b];
    c[b] = cb; S += cb;
  }
  sScanU[tid] = S;
  __syncthreads();
  for (int off = 1; off < NTHREADS; off <<= 1) {   // inclusive SUFFIX scan
    uint32_t add = (tid + off < NTHREADS) ? sScanU[tid + off] : 0u;
    __syncthreads();
    sScanU[tid] += add;
    __syncthreads();
  }
  uint32_t inclSuf = sScanU[tid];        // sum over threads >= tid
  uint32_t exclSuf = inclSuf - S;        // sum over threads >  tid (exact, uint)
  if (exclSuf < (uint32_t)K && inclSuf >= (uint32_t)K) {  // unique owner
    uint32_t acc = exclSuf;
    #pragma unroll
    for (int b = 7; b >= 0; --b) {
      if (acc + c[b] >= (uint32_t)K) { *sBin = (uint32_t)(tid * 8 + b);
                                       *sKrem = K - (int)acc; break; }
      acc += c[b];
    }
  }
  __syncthreads();
}

__global__ __launch_bounds__(NTHREADS)
void Sampler_90314572301058_kernel(const float* __restrict__ logits,
                                   const float* __restrict__ temps,
                                   const float* __restrict__ uvar,
                                   const int*   __restrict__ topk,
                                   int*         __restrict__ out) {
  __shared__ uint32_t hist[NREP][4096];  // 32 KB, wave-alternating replicas
  __shared__ float    tile[2][TILE_EL];  // 16 KB: async staging, then cand list
  __shared__ float    sScan[NTHREADS];   //  2 KB
  __shared__ uint32_t sScanU[NTHREADS];  //  2 KB
  __shared__ float    sred[16];
  __shared__ uint32_t sB0, sB1, sB2;
  __shared__ int      sK1, sK2, sOwner, sCandCnt;

  const int row = blockIdx.x;
  const int tid = threadIdx.x;
  const int rep = (tid >> 5) & (NREP - 1);
  const float* rowPtr = logits + (size_t)row * VV;
  uint32_t* histFlat = &hist[0][0];
  uint32_t* cand = (uint32_t*)&tile[0][0];   // alias: staging buf is free later

  int K = topk[0];
  if (K < 1) K = 1;
  if (K > VV) K = VV;
  const float T    = temps[row];
  const float invT = 1.0f / T;
  const float uu   = uvar[row];

  if (tid == 0) { sOwner = INT_MAX; sCandCnt = 0; }
  for (int i = tid; i < NREP * 4096; i += NTHREADS) histFlat[i] = 0u;
  __syncthreads();

  // ---------------- Pass 1 (HBM): hist on ord[31:20] + row max --------------
  // Async global->LDS staging (gfx1250 async data mover), double-buffered;
  // each lane consumes only the 16 bytes it requested -> only per-wave
  // s_wait_asynccnt needed, no block barrier in the pipeline.
  const uint32_t ldsOff0 = (uint32_t)(uintptr_t)&tile[0][tid * 4];
  const uint32_t ldsOff1 = (uint32_t)(uintptr_t)&tile[1][tid * 4];
  float lmax = -__builtin_inff();
  {
    uint64_t ga0 = (uint64_t)(uintptr_t)(rowPtr + tid * 4);
    asm volatile("global_load_async_to_lds_b128 %0, %1, off"
                 :: "v"(ldsOff0), "v"(ga0) : "memory");
    for (int t = 0; t < NTILES; ++t) {
      if (t + 1 < NTILES) {
        uint64_t ga = (uint64_t)(uintptr_t)(rowPtr + (t + 1) * TILE_EL + tid * 4);
        uint32_t lo = ((t + 1) & 1) ? ldsOff1 : ldsOff0;
        asm volatile("global_load_async_to_lds_b128 %0, %1, off"
                     :: "v"(lo), "v"(ga) : "memory");
        asm volatile("s_wait_asynccnt 0x1" ::: "memory");
      } else {
        asm volatile("s_wait_asynccnt 0x0" ::: "memory");
      }
      const float* p = &tile[t & 1][tid * 4];
      #pragma unroll
      for (int e = 0; e < 4; ++e) {
        float x = p[e];
        lmax = fmaxf(lmax, x);
        atomicAdd(&hist[rep][f2ord(x) >> 20], 1u);
      }
    }
    float x = rowPtr[TAIL_BASE + tid];            // 512-element tail
    lmax = fmaxf(lmax, x);
    atomicAdd(&hist[rep][f2ord(x) >> 20], 1u);
  }
  __syncthreads();
  const float rowMax = blockMax(lmax, sred);

  selectBin(hist, K, sScanU, &sB0, &sK1);
  const uint32_t b0 = sB0; const int K1 = sK1;
  __syncthreads();
  for (int i = tid; i < NREP * 4096; i += NTHREADS) histFlat[i] = 0u;
  __syncthreads();

  // ------- Pass 2 (L2): hist ord[19:8] within b0 + compact candidates -------
  // Expected candidates for the far-tail bin: O(10..100) of 152064 -> the
  // final byte is resolved from LDS, skipping a third 77.8 MB sweep.
  {
    const float4* r4 = (const float4*)rowPtr;
    for (int j = tid; j < VV / 4; j += NTHREADS) {
      float4 v = r4[j];
      float xs[4] = {v.x, v.y, v.z, v.w};
      #pragma unroll
      for (int e = 0; e < 4; ++e) {
        uint32_t o = f2ord(xs[e]);
        if ((o >> 20) == b0) {
          atomicAdd(&hist[rep][(o >> 8) & 0xFFFu], 1u);
          int idx = atomicAdd(&sCandCnt, 1);
          if (idx < CAND_CAP) cand[idx] = o;
        }
      }
    }
  }
  __syncthreads();
  const int nCand = sCandCnt;
  selectBin(hist, K1, sScanU, &sB1, &sK2);
  const uint32_t b1 = sB1; const int K2 = sK2;
  const uint32_t p01 = (b0 << 12) | b1;
  __syncthreads();
  for (int i = tid; i < NREP * 4096; i += NTHREADS) histFlat[i] = 0u;
  __syncthreads();

  // -------- Pass 3: hist on ord[7:0] within prefix p01 (LDS fast path) ------
  if (nCand <= CAND_CAP) {
    for (int i = tid; i < nCand; i += NTHREADS) {
      uint32_t o = cand[i];
      if ((o >> 8) == p01) atomicAdd(&hist[rep][o & 0xFFu], 1u);
    }
  } else {                                        // overflow fallback (rare)
    const float4* r4 = (const float4*)rowPtr;
    for (int j = tid; j < VV / 4; j += NTHREADS) {
      float4 v = r4[j];
      float xs[4] = {v.x, v.y, v.z, v.w};
      #pragma unroll
      for (int e = 0; e < 4; ++e) {
        uint32_t o = f2ord(xs[e]);
        if ((o >> 8) == p01) atomicAdd(&hist[rep][o & 0xFFu], 1u);
      }
    }
  }
  __syncthreads();
  selectBin(hist, K2, sScanU, &sB2, &sK2);   // bins >=256 are zero -> works
  const uint32_t ordT = (b0 << 20) | (b1 << 12) | sB2;  // exact k-th largest

  // ------- Pass 4 (L2): chunk exp-sums + block scan (Z = scan total) -------
  const float Ms = rowMax * invT;
  const int base = tid * CHUNK;
  float lsum = 0.0f;
  for (int i = 0; i < CHUNK; ++i) {
    float x = rowPtr[base + i];
    if (f2ord(x) >= ordT) lsum += __expf(x * invT - Ms);
  }
  sScan[tid] = lsum;
  __syncthreads();
  for (int off = 1; off < NTHREADS; off <<= 1) {   // Hillis-Steele inclusive
    float add = (tid >= off) ? sScan[tid - off] : 0.0f;
    __syncthreads();
    sScan[tid] += add;
    __syncthreads();
  }
  const float Z = sScan[NTHREADS - 1];             // softmax denominator
  const float target = uu * Z;
  const float incl = sScan[tid];
  const float pre  = (tid == 0) ? 0.0f : sScan[tid - 1];
  if (pre < target && incl >= target) atomicMin(&sOwner, tid);
  __syncthreads();

  if (tid == sOwner) {
    float run = pre;
    int token = -1, lastKept = base + CHUNK - 1;
    for (int i = 0; i < CHUNK; ++i) {
      float x = rowPtr[base + i];
      if (f2ord(x) >= ordT) {
        lastKept = base + i;
        run += __expf(x * invT - Ms);
        if (run >= target) { token = base + i; break; }
      }
    }
    if (token < 0) token = lastKept;               // rounding fallback
    out[row] = token < (VV - 1) ? token : (VV - 1);
  }
  if (tid == 0 && sOwner == INT_MAX) {
    out[row] = (target <= 0.0f) ? 0 : (VV - 1);    // u<=0 or u*Z > total
  }
}

extern "C" void kernel_launch(void* const* d_in, const int* in_sizes, int n_in,
                              void* d_out, int out_size, void* d_ws, size_t ws_size,
                              hipStream_t stream) {
  (void)in_sizes; (void)n_in; (void)d_ws; (void)ws_size; (void)out_size;
  const float* logits = (const float*)d_in[0];
  const float* temps  = (const float*)d_in[1];
  const float* uvar   = (const float*)d_in[2];
  const int*   topk   = (const int*)d_in[3];
  int* out = (int*)d_out;
  Sampler_90314572301058_kernel<<<BROWS, NTHREADS, 0, stream>>>(
      logits, temps, uvar, topk, out);
}